// decompress_jpeg_5128190951849
// MI455X (gfx1250) — compile-verified
//
#include <hip/hip_runtime.h>

typedef float v2f __attribute__((ext_vector_type(2)));
typedef float v4f __attribute__((ext_vector_type(4)));
typedef float v8f __attribute__((ext_vector_type(8)));

// ---------------- compile-time IDCT basis generation ----------------
constexpr double kPI = 3.14159265358979323846;

constexpr double tcos(double x) {              // Taylor, |x| <= pi/2
  double x2 = x * x, term = 1.0, sum = 1.0;
  for (int k = 1; k <= 12; ++k) { term *= -x2 / double((2 * k - 1) * (2 * k)); sum += term; }
  return sum;
}
constexpr double cos16(int t) {                // cos(t * pi / 16)
  int m = t % 32; if (m < 0) m += 32;
  double s = 1.0;
  if (m > 16) m = 32 - m;
  if (m > 8) { m = 16 - m; s = -1.0; }
  return s * tcos(m * kPI / 16.0);
}

constexpr int QY[8][8] = {
  {16, 11, 10, 16, 24, 40, 51, 61},  {12, 12, 14, 19, 26, 58, 60, 55},
  {14, 13, 16, 24, 40, 57, 69, 56},  {14, 17, 22, 29, 51, 87, 80, 62},
  {18, 22, 37, 56, 68, 109, 103, 77},{24, 35, 55, 64, 81, 104, 113, 92},
  {49, 64, 78, 87, 103, 121, 120, 101},{72, 92, 95, 98, 112, 100, 103, 99}};
constexpr int QC4[4][4] = {{17,18,24,47},{18,21,26,66},{24,26,56,99},{47,66,99,99}};

// Transposed basis: Bt[j=(u*8+v)][k=(x*8+y)] =
//   0.25*quant[x,y]*alpha[x,y]*cos((2u+1)x pi/16)*cos((2v+1)y pi/16)
// Row stride 68 floats: one ds_load_b64 per B fragment, bank-conflict-free
// ((4j + rk) mod 64 disjoint across the 32 lanes of a wave).
struct alignas(16) BTab { float v[64][68]; };

constexpr BTab makeB(bool luma) {
  BTab t{};
  double ct[8][8] = {};                        // ct[freq][spatial]
  for (int f = 0; f < 8; ++f)
    for (int s = 0; s < 8; ++s) ct[f][s] = cos16((2 * s + 1) * f);
  for (int x = 0; x < 8; ++x)
    for (int y = 0; y < 8; ++y) {
      double q  = luma ? double(QY[x][y]) : ((x < 4 && y < 4) ? double(QC4[x][y]) : 99.0);
      double ax = (x == 0) ? 0.70710678118654752440 : 1.0;
      double ay = (y == 0) ? 0.70710678118654752440 : 1.0;
      double sc = 0.25 * q * ax * ay;
      for (int u = 0; u < 8; ++u)
        for (int v = 0; v < 8; ++v)
          t.v[u * 8 + v][x * 8 + y] = (float)(sc * ct[x][u] * ct[y][v]);
    }
  return t;
}

__constant__ const BTab BT_Y = makeB(true);
__constant__ const BTab BT_C = makeB(false);

#define TPB 192

// One workgroup (6 wave32) decodes one 64x64 pixel tile of one image:
// waves 0-3: 16 Y blocks each (WMMA M-tile), wave 4: 16 Cb, wave 5: 16 Cr.
__global__ __launch_bounds__(TPB) void jpeg_decompress_kernel(
    const float* __restrict__ ycoef, const float* __restrict__ cbcoef,
    const float* __restrict__ crcoef, float* __restrict__ out)
{
  __shared__ float sB[2 * 64 * 68];   // transposed Y basis, C basis
  __shared__ float sY[64 * 68];       // luma spatial tile (stride 68)
  __shared__ float sC[2 * 32 * 36];   // cb, cr spatial tiles (stride 36)

  const int tid  = threadIdx.x;
  const int b    = blockIdx.y;        // batch image
  const int tile = blockIdx.x;        // 8x8 tiles of 64x64 px
  const int ty   = tile >> 3, tx = tile & 7;

  // ---- stage IDCT basis matrices into LDS with b128 copies ----
  {
    const v4f* srcY = (const v4f*)BT_Y.v;
    const v4f* srcC = (const v4f*)BT_C.v;
    v4f* dst = (v4f*)sB;
    for (int i = tid; i < 2 * 1088; i += TPB)
      dst[i] = (i < 1088) ? srcY[i] : srcC[i - 1088];
  }
  __syncthreads();

  const int wave = tid >> 5;
  const int lane = tid & 31;
  const int h    = lane >> 4;         // K-half (ISA A/B fragment layout)
  const int mi   = lane & 15;         // A-row (block) / B-column index

  // ---- per-lane A source pointer: 16 DCT blocks per wave ----
  const float* aptr;
  const float* bp;                    // per-lane B fragment base (imm offsets only)
  if (wave < 4) {                     // luma: wave covers block-rows 2w,2w+1 x 8 cols
    int n = (ty * 8 + 2 * wave + (mi >> 3)) * 64 + (tx * 8 + (mi & 7));
    aptr  = ycoef + (((size_t)b * 4096 + n) << 6) + (h << 1);
    bp    = sB + mi * 68 + (h << 1);
  } else {                            // chroma: 4x4 blocks of the 32x32 chroma tile
    const float* src = (wave == 4) ? cbcoef : crcoef;
    int n = (ty * 4 + (mi >> 2)) * 32 + (tx * 4 + (mi & 3));
    aptr  = src + (((size_t)b * 1024 + n) << 6) + (h << 1);
    bp    = sB + 64 * 68 + mi * 68 + (h << 1);
  }

  // ---- WMMA GEMM: D[16x64] = A[16 blk x 64 coeff] * B[64x64 basis] + 128 ----
  v8f acc[4];
  #pragma unroll
  for (int nt = 0; nt < 4; ++nt)
    #pragma unroll
    for (int r = 0; r < 8; ++r) acc[nt][r] = 128.0f;

  #pragma unroll
  for (int kk = 0; kk < 16; ++kk) {
    v2f a = *(const v2f*)(aptr + (kk << 2));              // A: K = 4kk+2h, +1
    #pragma unroll
    for (int nt = 0; nt < 4; ++nt) {
      // B fragment: Bt[j = nt*16+mi][rk = 4kk+2h .. +1] -> single ds_load_b64
      v2f bf = *(const v2f*)(bp + nt * (16 * 68) + (kk << 2));
      acc[nt] = __builtin_amdgcn_wmma_f32_16x16x4_f32(
          false, a, false, bf, (short)0, acc[nt], false, false);
    }
  }

  // ---- scatter D fragments (row M=r+8h, col j=nt*16+mi) into spatial LDS ----
  if (wave < 4) {
    #pragma unroll
    for (int nt = 0; nt < 4; ++nt) {
      int j = nt * 16 + mi, u = j >> 3, v = j & 7;
      float* row = sY + ((2 * wave + h) * 8 + u) * 68 + v; // m>>3==h, m&7==r
      #pragma unroll
      for (int r = 0; r < 8; ++r) row[r * 8] = acc[nt][r];
    }
  } else {
    float* plane = sC + (wave - 4) * (32 * 36);
    #pragma unroll
    for (int nt = 0; nt < 4; ++nt) {
      int j = nt * 16 + mi, u = j >> 3, v = j & 7;
      #pragma unroll
      for (int r = 0; r < 8; ++r) {
        int m = r + (h << 3);
        plane[((m >> 2) * 8 + u) * 36 + (m & 3) * 8 + v] = acc[nt][r];
      }
    }
  }
  __syncthreads();

  // ---- upsample + YCbCr->RGB + clamp, b128 nontemporal streaming stores ----
  const size_t HW = 512 * 512;
  float* outb = out + (size_t)b * 3 * HW + (size_t)(ty * 64) * 512 + tx * 64;
  #pragma unroll
  for (int it = 0; it < 6; ++it) {
    int q = tid + it * TPB;                         // 1024 quads of 4 px
    if (q < 1024) {
      int row = q >> 4, col = (q & 15) << 2;
      v4f yv = *(const v4f*)&sY[row * 68 + col];
      int co = (row >> 1) * 36 + (col >> 1);
      v2f cbv = *(const v2f*)&sC[co];
      v2f crv = *(const v2f*)&sC[32 * 36 + co];
      v4f R, G, B;
      #pragma unroll
      for (int i = 0; i < 4; ++i) {
        float cb = cbv[i >> 1] - 128.0f;
        float cr = crv[i >> 1] - 128.0f;
        float yy = yv[i];
        float r = yy + 1.402f * cr;
        float g = yy - 0.344136f * cb - 0.714136f * cr;
        float bl = yy + 1.772f * cb;
        R[i] = fminf(fmaxf(r, 0.0f), 255.0f) * (1.0f / 255.0f);
        G[i] = fminf(fmaxf(g, 0.0f), 255.0f) * (1.0f / 255.0f);
        B[i] = fminf(fmaxf(bl, 0.0f), 255.0f) * (1.0f / 255.0f);
      }
      float* p = outb + (size_t)row * 512 + col;
      __builtin_nontemporal_store(R, (v4f*)p);
      __builtin_nontemporal_store(G, (v4f*)(p + HW));
      __builtin_nontemporal_store(B, (v4f*)(p + 2 * HW));
    }
  }
}

extern "C" void kernel_launch(void* const* d_in, const int* in_sizes, int n_in,
                              void* d_out, int out_size, void* d_ws, size_t ws_size,
                              hipStream_t stream) {
  const float* y  = (const float*)d_in[0];   // [B, 4096, 8, 8]
  const float* cb = (const float*)d_in[1];   // [B, 1024, 8, 8]
  const float* cr = (const float*)d_in[2];   // [B, 1024, 8, 8]
  int B = in_sizes[0] >> 18;                 // elems / (4096*64)
  dim3 grid(64, (unsigned)B);                // 64 tiles per image
  jpeg_decompress_kernel<<<grid, TPB, 0, stream>>>(y, cb, cr, (float*)d_out);
}